// MultiSimilarityLoss_65231963292299
// MI455X (gfx1250) — compile-verified
//
#include <hip/hip_runtime.h>
#include <hip/hip_bf16.h>

// Multi-similarity loss, fused two-pass fp32-WMMA implementation for gfx1250.
//
// B=20480, D=512, GROUP=5, N=B/GROUP=4096.
// sim = f @ inputs^T (f = inputs[0::5]) is a 4096x20480x512 fp32 GEMM (~86 GFLOP).
// inputs (42 MB) is L2-resident (192 MB L2), so the kernel is matrix-pipe bound:
// fp32 V_WMMA_16X16X4_F32 is the numerically required path (sim ~ N(0,512) feeds
// exp(10*(sim-1)); bf16's 2^-8 relative error would corrupt the exp terms).
//
// Pass 1: per-row neg_max (excluding own 5-col block) + extract 4 positives.
// Pass 2: recompute sim tiles, masked exp-sums -> per-block partials.
// Finalize: deterministic single-thread sum of 256 block partials -> 3 scalars.
//
// Per wave: 16x64 macro-tile = 4 independent WMMA accumulator chains (ILP to
// hide XDL latency), A fragment ds_load amortized 4x. A tile staged to LDS by
// the Tensor Data Mover (tensor_load_to_lds + s_wait_tensorcnt) with TDM
// padding (2 DWORDs per 256-DWORD chunk) so A-fragment LDS reads are
// bank-conflict-free (chunk stride 258 floats -> banks 4*m apart).

#define EP     0.1f
#define BETA   10.0f
#define ALPHA  2.0f
#define LMDA   1.0f
#define GROUP  5
#define DDIM   512

typedef __attribute__((ext_vector_type(2))) float v2f;
typedef __attribute__((ext_vector_type(8))) float v8f;
typedef __attribute__((ext_vector_type(4))) unsigned int u32x4;
typedef __attribute__((ext_vector_type(8))) int i32x8;
typedef __attribute__((ext_vector_type(4))) int i32x4;

// LDS A layout: row m stored as two 256-float chunks, each followed by 2 pad
// floats (TDM pad_interval=256 DW, pad_amount=2 DW). Chunk ci = m*2 + k/256,
// float offset = ci*258 + (k%256). 16 rows -> 8256 floats (33 KB).
#define ACHUNK 258
#define ALDS   (16 * 2 * ACHUNK)

// ---- stage 16 f-rows (inputs rows 5*(rowBase+m)) into LDS ------------------
__device__ __forceinline__ void load_A_tile(const float* __restrict__ inp,
                                            int rowBase, float* lds) {
#if __has_builtin(__builtin_amdgcn_tensor_load_to_lds)
    if (threadIdx.x < 32) {   // one wave issues the TDM descriptor
        const unsigned long long ga =
            (unsigned long long)(uintptr_t)(inp + (size_t)(GROUP * rowBase) * DDIM);
        const unsigned int lds_off =
            (unsigned int)((uintptr_t)lds & 0xffffffffu);   // LDS aperture: low 32 bits = LDS offset

        u32x4 g0;
        g0[0] = 1u;                                          // count=1, user descriptor
        g0[1] = lds_off;                                     // lds_addr
        g0[2] = (unsigned int)(ga & 0xffffffffu);            // global_addr[31:0]
        g0[3] = (unsigned int)((ga >> 32) & 0x1ffffffu)      // global_addr[56:32]
              | (2u << 30);                                  // type=2 ("image")

        i32x8 g1;
        g1[0] = (int)((2u << 16)        // data_size = 4 bytes
              | (1u << 20)              // pad_enable
              | (7u << 22)              // pad_interval: 256 DWORDs
              | (1u << 25));            // pad_amount: 2 DWORDs
        g1[1] = (int)(512u << 16);      // tensor_dim0 = 512 elements (bits 79:48)
        g1[2] = (int)(16u << 16);       // tensor_dim1 = 16 lines   (bits 111:80)
        g1[3] = (int)(512u << 16);      // tile_dim0 = 512          (bits 127:112)
        g1[4] = (int)(16u);             // tile_dim1 = 16, tile_dim2 = 0
        g1[5] = (int)(GROUP * DDIM);    // tensor_dim0_stride = 2560 elements
        g1[6] = 0;                      // stride[47:32]=0, dim1_stride lo (unused, 2D)
        g1[7] = 0;

        i32x4 z4 = {0, 0, 0, 0};                 // groups 2/3 unused (2D tensor)
        i32x8 z8 = {0, 0, 0, 0, 0, 0, 0, 0};     // extra group (clang-23 6-arg form)
        __builtin_amdgcn_tensor_load_to_lds(g0, g1, z4, z4, z8, 0);
        __builtin_amdgcn_s_wait_tensorcnt(0);
    }
#else
    for (int idx = threadIdx.x; idx < 16 * DDIM; idx += blockDim.x) {
        const int m = idx >> 9;
        const int k = idx & (DDIM - 1);
        lds[(m * 2 + (k >> 8)) * ACHUNK + (k & 255)] =
            inp[(size_t)(GROUP * (rowBase + m)) * DDIM + k];
    }
#endif
}

// ---- 16x64 sim macro-tile: 4 col-tiles, 4 independent WMMA chains ----------
// A frag (16x4 f32): lanes 0-15 M=lane, K={k0,k0+1}; lanes 16-31 K={k0+2,k0+3}.
// B frag (4x16 f32): lane holds column n = lane&15, same K-half split.
// C/D (16x16 f32): VGPR r -> row r (lanes 0-15) / r+8 (lanes 16-31), col = lane&15.
__device__ __forceinline__ void sim_tile4(const float* lds,
                                          const float* __restrict__ inp,
                                          int c0, int lane, v8f* acc) {
    const int m  = lane & 15;
    const int kh = (lane >> 4) << 1;                       // 0 or 2
    const float* b0 = inp + (size_t)(c0 + m) * DDIM + kh;
#pragma unroll
    for (int half = 0; half < 2; ++half) {                 // K split at 256 (TDM pad chunk)
        const float* arow = lds + (m * 2 + half) * ACHUNK + kh;
        const float* bh   = b0 + half * 256;
#pragma unroll 4
        for (int k0 = 0; k0 < 256; k0 += 4) {
            v2f a = { arow[k0], arow[k0 + 1] };
#pragma unroll
            for (int ct = 0; ct < 4; ++ct) {
                const float* b = bh + (size_t)ct * 16 * DDIM;
                v2f bv = { b[k0], b[k0 + 1] };
                acc[ct] = __builtin_amdgcn_wmma_f32_16x16x4_f32(
                              false, a, false, bv, (short)0, acc[ct], false, false);
            }
        }
    }
}

// =============================== PASS 1 =====================================
__global__ void __launch_bounds__(256)
msloss_pass1(const float* __restrict__ inp, float* __restrict__ pos,
             float* __restrict__ negmax, int Bsz) {
    __shared__ float Af[ALDS];
    __shared__ float wmax[8][16];

    const int rowBase = blockIdx.x * 16;
    load_A_tile(inp, rowBase, Af);
    __syncthreads();

    const int wave    = threadIdx.x >> 5;
    const int lane    = threadIdx.x & 31;
    const int halfsel = lane >> 4;
    const int nsup    = Bsz >> 6;          // 320 super-tiles of 64 cols, % 8 == 0

    float rmax[8];
#pragma unroll
    for (int r = 0; r < 8; ++r) rmax[r] = -3.402823466e38f;

    for (int s = wave; s < nsup; s += 8) {
        const int c0 = s << 6;
        if (s + 8 < nsup) {                // prefetch next macro-tile columns
            const int cn = (s + 8) << 6;
            __builtin_prefetch(inp + (size_t)(cn + lane) * DDIM, 0, 3);
            __builtin_prefetch(inp + (size_t)(cn + 32 + lane) * DDIM, 0, 3);
        }

        v8f acc[4];
#pragma unroll
        for (int ct = 0; ct < 4; ++ct) acc[ct] = (v8f){};
        sim_tile4(Af, inp, c0, lane, acc);

#pragma unroll
        for (int ct = 0; ct < 4; ++ct) {
            const int j = c0 + ct * 16 + (lane & 15);
#pragma unroll
            for (int r = 0; r < 8; ++r) {
                const int   gm = rowBase + r + 8 * halfsel;   // global row
                const int   d0 = GROUP * gm;                  // own-block first column
                const float v  = acc[ct][r];
                if (j > d0 && j <= d0 + (GROUP - 1)) {
                    pos[gm * 4 + (j - d0 - 1)] = v;           // one of the 4 positives
                } else if (j != d0) {
                    rmax[r] = fmaxf(rmax[r], v);              // negative candidate
                }
            }
        }
    }

    // reduce max across the 16-lane halves (rows live per half)
#pragma unroll
    for (int off = 1; off < 16; off <<= 1)
#pragma unroll
        for (int r = 0; r < 8; ++r)
            rmax[r] = fmaxf(rmax[r], __shfl_xor(rmax[r], off, 32));

    if ((lane & 15) == 0) {
        const int base = halfsel * 8;
#pragma unroll
        for (int r = 0; r < 8; ++r) wmax[wave][base + r] = rmax[r];
    }
    __syncthreads();

    if (threadIdx.x < 16) {
        float m = wmax[0][threadIdx.x];
#pragma unroll
        for (int w = 1; w < 8; ++w) m = fmaxf(m, wmax[w][threadIdx.x]);
        negmax[rowBase + threadIdx.x] = m;
    }
}

// =============================== PASS 2 =====================================
__global__ void __launch_bounds__(256)
msloss_pass2(const float* __restrict__ inp, const float* __restrict__ pos,
             const float* __restrict__ negmax, float* __restrict__ partials,
             int Bsz) {
    __shared__ float Af[ALDS];
    __shared__ float wsum[8][16];
    __shared__ float wcnt[8][16];
    __shared__ float pmin_s[16];
    __shared__ float rloss[16], rpc[16], rnc[16];

    const int rowBase = blockIdx.x * 16;
    load_A_tile(inp, rowBase, Af);
    if (threadIdx.x < 16) {
        const int gi = rowBase + threadIdx.x;
        const float p0 = pos[gi * 4 + 0], p1 = pos[gi * 4 + 1];
        const float p2 = pos[gi * 4 + 2], p3 = pos[gi * 4 + 3];
        pmin_s[threadIdx.x] = fminf(fminf(p0, p1), fminf(p2, p3));
    }
    __syncthreads();

    const int wave    = threadIdx.x >> 5;
    const int lane    = threadIdx.x & 31;
    const int halfsel = lane >> 4;
    const int nsup    = Bsz >> 6;

    float nsum[8], ncnt[8];
#pragma unroll
    for (int r = 0; r < 8; ++r) { nsum[r] = 0.0f; ncnt[r] = 0.0f; }

    for (int s = wave; s < nsup; s += 8) {
        const int c0 = s << 6;
        if (s + 8 < nsup) {
            const int cn = (s + 8) << 6;
            __builtin_prefetch(inp + (size_t)(cn + lane) * DDIM, 0, 3);
            __builtin_prefetch(inp + (size_t)(cn + 32 + lane) * DDIM, 0, 3);
        }

        v8f acc[4];
#pragma unroll
        for (int ct = 0; ct < 4; ++ct) acc[ct] = (v8f){};
        sim_tile4(Af, inp, c0, lane, acc);

#pragma unroll
        for (int ct = 0; ct < 4; ++ct) {
            const int j = c0 + ct * 16 + (lane & 15);
#pragma unroll
            for (int r = 0; r < 8; ++r) {
                const int   lm = r + 8 * halfsel;       // local row
                const int   gm = rowBase + lm;
                const int   d0 = GROUP * gm;
                const float v  = acc[ct][r];
                const bool  own = (j >= d0) && (j <= d0 + (GROUP - 1));
                if (!own && v > pmin_s[lm] - EP) {
                    nsum[r] += expf(BETA * (v - LMDA));
                    ncnt[r] += 1.0f;
                }
            }
        }
    }

#pragma unroll
    for (int off = 1; off < 16; off <<= 1)
#pragma unroll
        for (int r = 0; r < 8; ++r) {
            nsum[r] += __shfl_xor(nsum[r], off, 32);
            ncnt[r] += __shfl_xor(ncnt[r], off, 32);
        }

    if ((lane & 15) == 0) {
        const int base = halfsel * 8;
#pragma unroll
        for (int r = 0; r < 8; ++r) {
            wsum[wave][base + r] = nsum[r];
            wcnt[wave][base + r] = ncnt[r];
        }
    }
    __syncthreads();

    if (threadIdx.x < 16) {
        const int gi = rowBase + threadIdx.x;
        float ns = 0.0f, nc = 0.0f;
#pragma unroll
        for (int w = 0; w < 8; ++w) { ns += wsum[w][threadIdx.x]; nc += wcnt[w][threadIdx.x]; }
        const float loss_n = log1pf(ns) / BETA;

        const float nm = negmax[gi];
        float ps = 0.0f, pc = 0.0f;
#pragma unroll
        for (int g = 0; g < 4; ++g) {
            const float p = pos[gi * 4 + g];
            if (p < nm + EP) { ps += expf(-ALPHA * (p - LMDA)); pc += 1.0f; }
        }
        const float loss_p = log1pf(ps) / ALPHA;

        rloss[threadIdx.x] = loss_p + loss_n;
        rpc[threadIdx.x]   = pc;
        rnc[threadIdx.x]   = nc;
    }
    __syncthreads();

    if (threadIdx.x == 0) {
        float L = 0.0f, P = 0.0f, Nn = 0.0f;
#pragma unroll
        for (int r = 0; r < 16; ++r) { L += rloss[r]; P += rpc[r]; Nn += rnc[r]; }
        partials[blockIdx.x * 3 + 0] = L;
        partials[blockIdx.x * 3 + 1] = P;
        partials[blockIdx.x * 3 + 2] = Nn;
    }
}

// ============================== FINALIZE ====================================
__global__ void msloss_finalize(const float* __restrict__ partials,
                                float* __restrict__ out, int nblocks, int Bsz) {
    float L = 0.0f, P = 0.0f, Nn = 0.0f;
    for (int i = 0; i < nblocks; ++i) {
        L  += partials[3 * i + 0];
        P  += partials[3 * i + 1];
        Nn += partials[3 * i + 2];
    }
    const int N = Bsz / GROUP;
    out[0] = L / (float)N;
    out[1] = P / (4.0f * (float)N);
    out[2] = Nn / ((float)(Bsz - GROUP) * (float)N);
}

extern "C" void kernel_launch(void* const* d_in, const int* in_sizes, int n_in,
                              void* d_out, int out_size, void* d_ws, size_t ws_size,
                              hipStream_t stream) {
    const float* inp = (const float*)d_in[0];   // (B, 512) fp32; targets (d_in[1]) unused
    const int Bsz = in_sizes[0] / DDIM;         // 20480
    const int N   = Bsz / GROUP;                // 4096
    const int nblocks = N / 16;                 // 256

    float* ws       = (float*)d_ws;
    float* pos      = ws;                       // N*4 floats
    float* negmax   = ws + (size_t)N * 4;       // N floats
    float* partials = negmax + N;               // nblocks*3 floats

    msloss_pass1<<<nblocks, 256, 0, stream>>>(inp, pos, negmax, Bsz);
    msloss_pass2<<<nblocks, 256, 0, stream>>>(inp, pos, negmax, partials, Bsz);
    msloss_finalize<<<1, 1, 0, stream>>>(partials, (float*)d_out, nblocks, Bsz);
}